// Memory_Network_61684320305314
// MI455X (gfx1250) — compile-verified
//
#include <hip/hip_runtime.h>
#include <cstdint>
#include <cstddef>

#define MEM_SIZE 262144
#define FEAT     512
#define TOPK     256
#define BATCH    128

#define MTILE    128   // key rows per workgroup

typedef __attribute__((ext_vector_type(2))) float v2f;
typedef __attribute__((ext_vector_type(4))) float v4f;
typedef __attribute__((ext_vector_type(8))) float v8f;

#if defined(__AMDGCN__) && __has_builtin(__builtin_amdgcn_global_load_async_to_lds_b128)
#define USE_ASYNC_LDS 1
#else
#define USE_ASYNC_LDS 0
#endif

#define AS1 __attribute__((address_space(1)))
#define AS3 __attribute__((address_space(3)))

typedef int v4i_vs __attribute__((vector_size(16)));   // matches builtin pointee type
typedef AS1 v4i_vs* gvec128_t;
typedef AS3 v4i_vs* lvec128_t;

template <int N>
__device__ __forceinline__ void wait_asynccnt() {
#if __has_builtin(__builtin_amdgcn_s_wait_asynccnt)
    __builtin_amdgcn_s_wait_asynccnt(N);
#else
    asm volatile("s_wait_asynccnt %0" ::"i"(N) : "memory");
#endif
}

#if USE_ASYNC_LDS
// async 16B copy: global -> LDS, tracked by ASYNCcnt
__device__ __forceinline__ void async_copy16(const float* g, float* l) {
    __builtin_amdgcn_global_load_async_to_lds_b128(
        (gvec128_t)(uintptr_t)g,
        (lvec128_t)(uint32_t)(uintptr_t)l,   // low 32 bits of generic LDS addr == LDS offset
        0, 0);
}
#endif

// ---------------------------------------------------------------------------
// Kernel 1: L2-normalize query rows  (128 rows x 512)
// ---------------------------------------------------------------------------
__global__ __launch_bounds__(256) void qnorm_kernel(const float* __restrict__ q,
                                                    float* __restrict__ qn) {
    const int b = blockIdx.x;
    const float* row = q + (size_t)b * FEAT;
    float ss = 0.f;
    for (int i = threadIdx.x; i < FEAT; i += 256) { float v = row[i]; ss += v * v; }
    for (int off = 16; off; off >>= 1) ss += __shfl_xor(ss, off, 32);
    __shared__ float red[8];
    const int lane = threadIdx.x & 31, wv = threadIdx.x >> 5;
    if (lane == 0) red[wv] = ss;
    __syncthreads();
    if (wv == 0) {
        float s = (lane < 8) ? red[lane] : 0.f;
        for (int off = 4; off; off >>= 1) s += __shfl_xor(s, off, 32);
        if (lane == 0) red[0] = rsqrtf(s);
    }
    __syncthreads();
    const float inv = red[0];
    for (int i = threadIdx.x; i < FEAT; i += 256)
        qn[(size_t)b * FEAT + i] = row[i] * inv;
}

// ---------------------------------------------------------------------------
// Kernel 2: cosine scores via V_WMMA_F32_16X16X4_F32
//   grid = MEM_SIZE/MTILE workgroups of 256 threads (8 waves)
//   per WG: scores[0..127][mBase .. mBase+127]
// ---------------------------------------------------------------------------
#if USE_ASYNC_LDS

#define KTILE 16
#define QP    20   // LDS pitch: 20*m mod 64 -> 16 distinct multiples of 4, conflict-free b64 reads
#define NT    (FEAT / KTILE)

__device__ __forceinline__ void stage_async(const float* __restrict__ gq,
                                            const float* __restrict__ gk,
                                            float (*qbuf)[QP], float (*kbuf)[QP],
                                            int tid, int mBase, int kt) {
    #pragma unroll
    for (int i = 0; i < 2; ++i) {
        const int qidx = tid + 256 * i;        // 0..511
        const int row  = qidx >> 2;            // 0..127
        const int c4   = (qidx & 3) * 4;       // 0,4,8,12
        async_copy16(gk + (size_t)(mBase + row) * FEAT + kt + c4, &kbuf[row][c4]);
        async_copy16(gq + (size_t)row * FEAT + kt + c4, &qbuf[row][c4]);
    }
}

__global__ __launch_bounds__(256) void score_wmma_kernel(const float* __restrict__ qn,
                                                         const float* __restrict__ keys,
                                                         float* __restrict__ scores) {
    __shared__ alignas(16) float qT[2][BATCH][QP];   // 20 KB
    __shared__ alignas(16) float kT[2][MTILE][QP];   // 20 KB

    const int tid   = threadIdx.x;
    const int lane  = tid & 31;
    const int wave  = tid >> 5;       // 0..7 : owns key columns [wave*16, wave*16+16)
    const int ln15  = lane & 15;
    const int grp   = lane >> 4;      // lane group -> K split per ISA A/B layout
    const int mBase = blockIdx.x * MTILE;

    v8f acc[8] = {};                  // 8 batch-tiles of 16 rows -> full 128 batch

    // prime the two-deep async pipeline
    stage_async(qn, keys, qT[0], kT[0], tid, mBase, 0);
    stage_async(qn, keys, qT[1], kT[1], tid, mBase, KTILE);

    for (int t = 0; t < NT; ++t) {
        // stage t complete once outstanding ops <= one stage (4 per thread)
        if (t + 1 < NT) wait_asynccnt<4>(); else wait_asynccnt<0>();
        __syncthreads();              // all waves' portions of tile t visible in LDS

        const float (*qb)[QP] = qT[t & 1];
        const float (*kb)[QP] = kT[t & 1];
        #pragma unroll
        for (int k4 = 0; k4 < KTILE; k4 += 4) {
            const int koff = k4 + 2 * grp;    // ISA: K = reg + 2*lanegroup -> contiguous pair
            const v2f bfrag = *(const v2f*)&kb[wave * 16 + ln15][koff];
            #pragma unroll
            for (int bt = 0; bt < 8; ++bt) {
                const v2f afrag = *(const v2f*)&qb[bt * 16 + ln15][koff];
                acc[bt] = __builtin_amdgcn_wmma_f32_16x16x4_f32(
                    false, afrag, false, bfrag, (short)0, acc[bt], false, false);
            }
        }
        __syncthreads();              // tile t fully consumed by all waves
        if (t + 2 < NT)               // refill the buffer just freed
            stage_async(qn, keys, qT[t & 1], kT[t & 1], tid, mBase, (t + 2) * KTILE);
    }

    // D layout: lanes 0-15 -> M=reg, lanes 16-31 -> M=reg+8 ; N = lane&15
    const int mcol = mBase + wave * 16 + ln15;
    #pragma unroll
    for (int bt = 0; bt < 8; ++bt) {
        #pragma unroll
        for (int r = 0; r < 8; ++r) {
            const int b = bt * 16 + r + 8 * grp;
            scores[(size_t)b * MEM_SIZE + mcol] = acc[bt][r];
        }
    }
}

#else  // ---------------- fallback: synchronous staging ----

#define KTILE 32
#define QP    36

__global__ __launch_bounds__(256) void score_wmma_kernel(const float* __restrict__ qn,
                                                         const float* __restrict__ keys,
                                                         float* __restrict__ scores) {
    __shared__ alignas(16) float qT[BATCH][QP];
    __shared__ alignas(16) float kT[MTILE][QP];

    const int tid   = threadIdx.x;
    const int lane  = tid & 31;
    const int wave  = tid >> 5;
    const int ln15  = lane & 15;
    const int grp   = lane >> 4;
    const int mBase = blockIdx.x * MTILE;

    v8f acc[8] = {};

    for (int kt = 0; kt < FEAT; kt += KTILE) {
        __syncthreads();
        {
            const int r  = tid >> 3;
            const int c4 = (tid & 7) * 4;
            #pragma unroll
            for (int i = 0; i < 4; ++i) {
                const int row = r + 32 * i;
                *(v4f*)&kT[row][c4] =
                    *(const v4f*)(keys + (size_t)(mBase + row) * FEAT + kt + c4);
                *(v4f*)&qT[row][c4] =
                    *(const v4f*)(qn + (size_t)row * FEAT + kt + c4);
            }
        }
        if (kt + KTILE < FEAT) {
            const int prow = tid & (MTILE - 1);
            __builtin_prefetch(keys + (size_t)(mBase + prow) * FEAT + kt + KTILE, 0, 0);
        }
        __syncthreads();

        #pragma unroll
        for (int k4 = 0; k4 < KTILE; k4 += 4) {
            const int koff = k4 + 2 * grp;
            const v2f bfrag = *(const v2f*)&kT[wave * 16 + ln15][koff];
            #pragma unroll
            for (int bt = 0; bt < 8; ++bt) {
                const v2f afrag = *(const v2f*)&qT[bt * 16 + ln15][koff];
                acc[bt] = __builtin_amdgcn_wmma_f32_16x16x4_f32(
                    false, afrag, false, bfrag, (short)0, acc[bt], false, false);
            }
        }
    }

    const int mcol = mBase + wave * 16 + ln15;
    #pragma unroll
    for (int bt = 0; bt < 8; ++bt) {
        #pragma unroll
        for (int r = 0; r < 8; ++r) {
            const int b = bt * 16 + r + 8 * grp;
            scores[(size_t)b * MEM_SIZE + mcol] = acc[bt][r];
        }
    }
}

#endif  // USE_ASYNC_LDS

// ---------------------------------------------------------------------------
// Kernel 3: per-row top-256 (radix-select + LDS bitonic sort)
// ---------------------------------------------------------------------------
#define NBINS 4096
#define CAND  4096

__device__ __forceinline__ unsigned f2ord(float f) {
    unsigned x = __float_as_uint(f);
    return (x & 0x80000000u) ? ~x : (x | 0x80000000u);
}

__global__ __launch_bounds__(1024) void topk_kernel(const float* __restrict__ scores,
                                                    float* __restrict__ outScore,
                                                    float* __restrict__ outIdxF,
                                                    int* __restrict__ idxBuf) {
    const int b = blockIdx.x;
    const float* row = scores + (size_t)b * MEM_SIZE;

    __shared__ unsigned hist[NBINS];
    __shared__ float    cScore[CAND];
    __shared__ int      cIdx[CAND];
    __shared__ int      sThreshBin;
    __shared__ unsigned sCnt;

    for (int i = threadIdx.x; i < NBINS; i += 1024) hist[i] = 0u;
    if (threadIdx.x == 0) sCnt = 0u;
    __syncthreads();

    for (int i = threadIdx.x; i < MEM_SIZE; i += 1024)
        atomicAdd(&hist[f2ord(row[i]) >> 20], 1u);
    __syncthreads();

    if (threadIdx.x == 0) {
        unsigned cum = 0; int tb = 0;
        for (int i = NBINS - 1; i >= 0; --i) {
            cum += hist[i];
            if (cum >= TOPK) { tb = i; break; }
        }
        sThreshBin = tb;
    }
    __syncthreads();
    const unsigned tb = (unsigned)sThreshBin;

    for (int i = threadIdx.x; i < MEM_SIZE; i += 1024) {
        const float v = row[i];
        if ((f2ord(v) >> 20) >= tb) {
            const unsigned p = atomicAdd(&sCnt, 1u);
            if (p < CAND) { cScore[p] = v; cIdx[p] = i; }
        }
    }
    __syncthreads();
    const unsigned nc = sCnt < (unsigned)CAND ? sCnt : (unsigned)CAND;
    for (int i = threadIdx.x; i < CAND; i += 1024)
        if ((unsigned)i >= nc) { cScore[i] = -__builtin_inff(); cIdx[i] = 0; }
    __syncthreads();

    for (int k = 2; k <= CAND; k <<= 1) {
        for (int j = k >> 1; j > 0; j >>= 1) {
            for (int t = threadIdx.x; t < CAND; t += 1024) {
                const int l = t ^ j;
                if (l > t) {
                    const bool desc = ((t & k) == 0);
                    const float a = cScore[t], c = cScore[l];
                    if (desc ? (a < c) : (a > c)) {
                        cScore[t] = c; cScore[l] = a;
                        const int ia = cIdx[t]; cIdx[t] = cIdx[l]; cIdx[l] = ia;
                    }
                }
            }
            __syncthreads();
        }
    }

    for (int i = threadIdx.x; i < TOPK; i += 1024) {
        outScore[(size_t)b * TOPK + i] = cScore[i];
        outIdxF [(size_t)b * TOPK + i] = (float)cIdx[i];
        idxBuf  [(size_t)b * TOPK + i] = cIdx[i];
    }
}

// ---------------------------------------------------------------------------
// Kernel 4: gather color_value rows for top-k indices
// ---------------------------------------------------------------------------
__global__ __launch_bounds__(128) void gather_kernel(const float* __restrict__ cv,
                                                     const int* __restrict__ idxBuf,
                                                     float* __restrict__ outFeat) {
    const int p = blockIdx.x;                   // b*TOPK + k
    const int m = idxBuf[p];
    const v4f* src = (const v4f*)(cv + (size_t)m * FEAT);
    v4f*       dst = (v4f*)(outFeat + (size_t)p * FEAT);
    dst[threadIdx.x] = src[threadIdx.x];
}

// ---------------------------------------------------------------------------
extern "C" void kernel_launch(void* const* d_in, const int* in_sizes, int n_in,
                              void* d_out, int out_size, void* d_ws, size_t ws_size,
                              hipStream_t stream) {
    const float* query = (const float*)d_in[0];   // [128, 512]
    const float* skey  = (const float*)d_in[1];   // [262144, 512]
    const float* cval  = (const float*)d_in[2];   // [262144, 512]

    float* ws     = (float*)d_ws;
    float* qn     = ws;                                            // 128*512
    float* scores = ws + (size_t)BATCH * FEAT;                     // 128*262144
    int*   idxBuf = (int*)(ws + (size_t)BATCH * FEAT + (size_t)BATCH * MEM_SIZE);

    float* out      = (float*)d_out;
    float* outFeat  = out;                                         // [128,256,512]
    float* outScore = out + (size_t)BATCH * TOPK * FEAT;           // [128,256]
    float* outIdxF  = outScore + (size_t)BATCH * TOPK;             // [128,256]

    qnorm_kernel<<<BATCH, 256, 0, stream>>>(query, qn);
    score_wmma_kernel<<<MEM_SIZE / MTILE, 256, 0, stream>>>(qn, skey, scores);
    topk_kernel<<<BATCH, 1024, 0, stream>>>(scores, outScore, outIdxF, idxBuf);
    gather_kernel<<<BATCH * TOPK, 128, 0, stream>>>(cval, idxBuf, outFeat);
}